// DeConv_40029095199190
// MI455X (gfx1250) — compile-verified
//
#include <hip/hip_runtime.h>

// ---------------- problem constants (from reference) ----------------
#define KINV    5
#define GC      16
#define GROUPS  8           // 128 / 16
#define CIN     256
#define COUT    128
#define CRED    32          // 128 / 4
#define CSPAN   200         // K*K*groups = 25*8
#define P64     4096        // 64*64 low-res pixels per image
#define W64     64
#define H64LOW  64
#define HFULL   128
#define WFULL   128
#define PFULL   16384       // 128*128
#define BATCH   4
#define EPS     1e-5f

typedef float v2f __attribute__((ext_vector_type(2)));
typedef float v8f __attribute__((ext_vector_type(8)));

// D(16x16,f32) = A(16x4,f32) x B(4x16,f32) + C   (codegen-confirmed round 1)
// A layout: lane<16 -> row M=lane, v0=K0,v1=K1 ; lane>=16 -> row M=lane-16, v0=K2,v1=K3
// B layout: lane<16 -> col N=lane, v0=K0,v1=K1 ; lane>=16 -> col N=lane-16, v0=K2,v1=K3
// C/D layout: VGPR r -> row (r + 8*(lane>=16)), col = lane&15
static __device__ __forceinline__ v8f wmma_f32_16x16x4(v2f a, v2f b, v8f c) {
  return __builtin_amdgcn_wmma_f32_16x16x4_f32(false, a, false, b, (short)0, c,
                                               false, false);
}

// ================= kernel A: h64 = W1(128x256) * x + b1 ==================
// one wave per 32(chan) x 16(pixel) tile: two M-tiles share one B operand
// (halves x re-reads vs one tile/wave; 2 independent WMMA chains per k-step)
__global__ __launch_bounds__(128) void k_conv1_wmma(
    const float* __restrict__ x,      // [4,256,64,64]
    const float* __restrict__ w,      // [128,256]
    const float* __restrict__ bias,   // [128]
    float* __restrict__ h64)          // [4,128,64,64]
{
  const int wave = (blockIdx.x * 128 + threadIdx.x) >> 5;
  const int lane = threadIdx.x & 31;
  const int half = lane >> 4;
  const int l    = lane & 15;

  int t = wave;                 // waves: b(4) * mp(4) * nt(256)
  const int nt = t & 255; t >>= 8;
  const int mp = t & 3;   t >>= 2;
  const int b  = t;
  const int mBase   = mp * 32;
  const int pixBase = nt * 16;

  v8f acc0, acc1;
#pragma unroll
  for (int r = 0; r < 8; ++r) {
    acc0[r] = bias[mBase + 8 * half + r];
    acc1[r] = bias[mBase + 16 + 8 * half + r];
  }

  const float* wrow0 = w + (mBase + l) * CIN;
  const float* wrow1 = w + (mBase + 16 + l) * CIN;
  const float* xcol  = x + (size_t)b * CIN * P64 + pixBase + l;

  for (int k0 = 0; k0 < CIN; k0 += 4) {
    if ((k0 & 31) == 0)   // prefetch ahead in the K direction
      __builtin_prefetch(xcol + (size_t)(k0 + 32) * P64, 0, 1);
    v2f bb;
    bb.x = xcol[(size_t)(k0 + 2 * half) * P64];
    bb.y = xcol[(size_t)(k0 + 2 * half + 1) * P64];
    v2f a0 = *(const v2f*)(wrow0 + k0 + 2 * half);
    v2f a1 = *(const v2f*)(wrow1 + k0 + 2 * half);
    acc0 = wmma_f32_16x16x4(a0, bb, acc0);
    acc1 = wmma_f32_16x16x4(a1, bb, acc1);
  }

  float* o0 = h64 + (size_t)(b * COUT + mBase + 8 * half) * P64 + pixBase + l;
  float* o1 = o0 + (size_t)16 * P64;
#pragma unroll
  for (int r = 0; r < 8; ++r) {
    o0[(size_t)r * P64] = acc0[r];
    o1[(size_t)r * P64] = acc1[r];
  }
}

// ====== kernel B1: r64 = relu(bn(W_red(32x128) * h64 + b_red)) (BN folded) ======
__global__ __launch_bounds__(128) void k_red_wmma(
    const float* __restrict__ h64,    // [4,128,64,64]
    const float* __restrict__ w,      // [32,128]
    const float* __restrict__ bias,   // [32]
    const float* __restrict__ gamma,
    const float* __restrict__ beta,
    const float* __restrict__ mean,
    const float* __restrict__ var,
    float* __restrict__ r64)          // [4,32,64,64]
{
  const int wave = (blockIdx.x * 128 + threadIdx.x) >> 5;
  const int lane = threadIdx.x & 31;
  const int half = lane >> 4;
  const int l    = lane & 15;

  int t = wave;                 // waves: b(4) * mt(2) * nt(256)
  const int nt = t & 255; t >>= 8;
  const int mt = t & 1;   t >>= 1;
  const int b  = t;
  const int mBase   = mt * 16;
  const int pixBase = nt * 16;

  const int mA    = mBase + l;                         // A-row this lane feeds
  const float sA  = gamma[mA] * rsqrtf(var[mA] + EPS); // fold BN scale into A

  v8f acc;
#pragma unroll
  for (int r = 0; r < 8; ++r) {
    const int m  = mBase + 8 * half + r;
    const float s = gamma[m] * rsqrtf(var[m] + EPS);
    acc[r] = s * (bias[m] - mean[m]) + beta[m];        // folded BN bias
  }

  const float* wrow = w + mA * COUT;
  const float* hcol = h64 + (size_t)b * COUT * P64 + pixBase + l;

  for (int k0 = 0; k0 < COUT; k0 += 4) {
    v2f a = *(const v2f*)(wrow + k0 + 2 * half);
    a.x *= sA; a.y *= sA;
    v2f bb;
    bb.x = hcol[(size_t)(k0 + 2 * half) * P64];
    bb.y = hcol[(size_t)(k0 + 2 * half + 1) * P64];
    acc = wmma_f32_16x16x4(a, bb, acc);
  }

  float* o = r64 + (size_t)(b * CRED + mBase + 8 * half) * P64 + pixBase + l;
#pragma unroll
  for (int r = 0; r < 8; ++r) o[(size_t)r * P64] = fmaxf(acc[r], 0.0f);
}

// ========= kernel B2: wgt64 = W_span(200x32) * r64 + b_span (M padded to 208) =========
__global__ __launch_bounds__(128) void k_span_wmma(
    const float* __restrict__ r64,    // [4,32,64,64]
    const float* __restrict__ w,      // [200,32]
    const float* __restrict__ bias,   // [200]
    float* __restrict__ wgt64)        // [4,200,64,64]
{
  const int wave = (blockIdx.x * 128 + threadIdx.x) >> 5;
  const int lane = threadIdx.x & 31;
  const int half = lane >> 4;
  const int l    = lane & 15;

  int t = wave;                 // waves: b(4) * mt(13) * nt(256)
  const int nt = t & 255; t >>= 8;
  const int mt = t % 13;
  const int b  = t / 13;
  const int mBase   = mt * 16;
  const int pixBase = nt * 16;

  const int  mA   = mBase + l;
  const bool mAok = (mA < CSPAN);

  v8f acc;
#pragma unroll
  for (int r = 0; r < 8; ++r) {
    const int m = mBase + 8 * half + r;
    acc[r] = (m < CSPAN) ? bias[m] : 0.0f;
  }

  const float* wrow = w + (mAok ? mA : 0) * CRED;
  const float* rcol = r64 + (size_t)b * CRED * P64 + pixBase + l;

  for (int k0 = 0; k0 < CRED; k0 += 4) {
    v2f a;
    if (mAok) a = *(const v2f*)(wrow + k0 + 2 * half);
    else      { a.x = 0.0f; a.y = 0.0f; }
    v2f bb;
    bb.x = rcol[(size_t)(k0 + 2 * half) * P64];
    bb.y = rcol[(size_t)(k0 + 2 * half + 1) * P64];
    acc = wmma_f32_16x16x4(a, bb, acc);
  }

  float* o = wgt64 + (size_t)(b * CSPAN + mBase + 8 * half) * P64 + pixBase + l;
#pragma unroll
  for (int r = 0; r < 8; ++r)
    if (mBase + 8 * half + r < CSPAN) o[(size_t)r * P64] = acc[r];
}

// ====================== kernel C: involution (5x5 -> 3x3, LDS-tiled) ======================
// Block = (b, g, yy): owns full-res rows Y=2yy, 2yy+1 for all 128 X and 16 gc.
// Stage h64 rows yy-1..yy+1 (16gc x 3 x 64 = 12KB) and wgt (25 x 64 = 6.25KB) in LDS,
// then each thread combines 25 per-pixel weights -> 3x3 (parity) and does 9 FMA x 16 gc.
__global__ __launch_bounds__(256) void k_involution(
    const float* __restrict__ h64,    // [4,128,64,64]
    const float* __restrict__ wgt64,  // [4,200,64,64]
    float* __restrict__ out)          // [4,128,128,128]
{
  __shared__ float h_tile[GC][3][W64];          // 12 KB
  __shared__ float wgt_tile[KINV * KINV][W64];  // 6.25 KB

  int blk = blockIdx.x;         // blocks: b(4) * g(8) * yy(64)
  const int yy = blk & 63; blk >>= 6;
  const int g  = blk & 7;  blk >>= 3;
  const int b  = blk;
  const int t  = threadIdx.x;

  // ---- cooperative LDS fill (coalesced 64-float rows) ----
  const float* hbase = h64 + (size_t)(b * COUT + g * GC) * P64;
  for (int i = t; i < GC * 3 * W64; i += 256) {
    const int col = i & 63;
    const int r   = (i >> 6) % 3;
    const int gc  = i / (3 * W64);
    const int sy  = yy - 1 + r;
    h_tile[gc][r][col] = (sy >= 0 && sy < H64LOW)
        ? hbase[(size_t)gc * P64 + sy * W64 + col] : 0.0f;
  }
  const float* wbase = wgt64 + (size_t)(b * CSPAN + g * 25) * P64 + yy * W64;
  for (int i = t; i < 25 * W64; i += 256) {
    const int col = i & 63;
    const int k   = i >> 6;
    wgt_tile[k][col] = wbase[(size_t)k * P64 + col];
  }
  __syncthreads();

  const int X  = t & 127;
  const int pY = t >> 7;              // Y = 2*yy + pY
  const int Y  = 2 * yy + pY;
  const int cx = X >> 1;
  const int pX = X & 1;

  // column combine: 5 taps -> 3 source columns (parity dependent)
  float colw[5][3];
#pragma unroll
  for (int dy = 0; dy < 5; ++dy) {
    const float w0 = wgt_tile[dy * 5 + 0][cx];
    const float w1 = wgt_tile[dy * 5 + 1][cx];
    const float w2 = wgt_tile[dy * 5 + 2][cx];
    const float w3 = wgt_tile[dy * 5 + 3][cx];
    const float w4 = wgt_tile[dy * 5 + 4][cx];
    if (pX == 0) { colw[dy][0] = w0 + w1; colw[dy][1] = w2 + w3; colw[dy][2] = w4; }
    else         { colw[dy][0] = w0;      colw[dy][1] = w1 + w2; colw[dy][2] = w3 + w4; }
  }
  // row combine: 5 -> 3
  float cw[3][3];
#pragma unroll
  for (int c = 0; c < 3; ++c) {
    if (pY == 0) { cw[0][c] = colw[0][c] + colw[1][c]; cw[1][c] = colw[2][c] + colw[3][c]; cw[2][c] = colw[4][c]; }
    else         { cw[0][c] = colw[0][c];              cw[1][c] = colw[1][c] + colw[2][c]; cw[2][c] = colw[3][c] + colw[4][c]; }
  }

  const bool rx0 = (cx > 0), rx2 = (cx < W64 - 1);
  float* ob = out + (size_t)(b * COUT + g * GC) * PFULL + Y * WFULL + X;

#pragma unroll 4
  for (int gc = 0; gc < GC; ++gc) {
    float acc = 0.0f;
#pragma unroll
    for (int r = 0; r < 3; ++r) {
#pragma unroll
      for (int c = 0; c < 3; ++c) {
        const bool cok = (c == 0) ? rx0 : (c == 2) ? rx2 : true;
        const float v = cok ? h_tile[gc][r][cx - 1 + c] : 0.0f;
        acc = fmaf(cw[r][c], v, acc);
      }
    }
    ob[(size_t)gc * PFULL] = acc;
  }
}

// ============================== host launcher ==============================
extern "C" void kernel_launch(void* const* d_in, const int* in_sizes, int n_in,
                              void* d_out, int out_size, void* d_ws, size_t ws_size,
                              hipStream_t stream) {
  const float* x      = (const float*)d_in[0];
  const float* w1x1   = (const float*)d_in[1];
  const float* b1x1   = (const float*)d_in[2];
  const float* w_red  = (const float*)d_in[3];
  const float* b_red  = (const float*)d_in[4];
  const float* gamma  = (const float*)d_in[5];
  const float* beta   = (const float*)d_in[6];
  const float* mean   = (const float*)d_in[7];
  const float* var    = (const float*)d_in[8];
  const float* w_span = (const float*)d_in[9];
  const float* b_span = (const float*)d_in[10];
  float* out = (float*)d_out;

  float* ws    = (float*)d_ws;
  float* h64   = ws;                                   // 4*128*4096 = 2,097,152 f
  float* r64   = h64 + (size_t)BATCH * COUT * P64;     // 4*32*4096  =   524,288 f
  float* wgt64 = r64 + (size_t)BATCH * CRED * P64;     // 4*200*4096 = 3,276,800 f

  // A: 4*4*256 = 4096 waves, 4 waves/block -> 1024 blocks
  k_conv1_wmma<<<1024, 128, 0, stream>>>(x, w1x1, b1x1, h64);
  // B1: 4*2*256 = 2048 waves -> 512 blocks
  k_red_wmma<<<512, 128, 0, stream>>>(h64, w_red, b_red, gamma, beta, mean, var, r64);
  // B2: 4*13*256 = 13312 waves -> 3328 blocks
  k_span_wmma<<<3328, 128, 0, stream>>>(r64, w_span, b_span, wgt64);
  // C: blocks = 4*8*64 = 2048, 256 threads each
  k_involution<<<2048, 256, 0, stream>>>(h64, wgt64, out);
}